// DeterminantHead_72971494359228
// MI455X (gfx1250) — compile-verified
//
#include <hip/hip_runtime.h>
#include <math.h>

typedef __attribute__((ext_vector_type(2))) float v2f;
typedef __attribute__((ext_vector_type(8))) float v8f;

#define N_EL     64
#define NSPIN    32
#define D_MODEL  256
#define N_ATOM   16
#define HPAD     260   // 256 + 4 pad: rows land in distinct banks for the WMMA A loads

__device__ __forceinline__ float gelu_exact(float x) {
    return 0.5f * x * (1.0f + erff(x * 0.70710678118654752f));
}

__global__ __launch_bounds__(256)
void det_head_kernel(const float* __restrict__ features,      // [B,64,256]
                     const float* __restrict__ ecoords,       // [B,64,3]
                     const float* __restrict__ atom_coords,   // [16,3]
                     const float* __restrict__ upW1, const float* __restrict__ upb1,
                     const float* __restrict__ upW2, const float* __restrict__ upb2,
                     const float* __restrict__ up_pi, const float* __restrict__ up_sg,
                     const float* __restrict__ dnW1, const float* __restrict__ dnb1,
                     const float* __restrict__ dnW2, const float* __restrict__ dnb2,
                     const float* __restrict__ dn_pi, const float* __restrict__ dn_sg,
                     float* __restrict__ out)
{
    __shared__ float hbuf[NSPIN][HPAD];     // gelu(f@W1+b1), one spin at a time
    __shared__ float orb[2][NSPIN][33];     // raw orbital matrices (env folded into logdet)
    __shared__ float lenv[2][NSPIN];        // log|env| per electron
    __shared__ float res[2];

    const int tid  = threadIdx.x;
    const int lane = tid & 31;
    const int wv   = tid >> 5;          // 8 waves
    const int b    = blockIdx.x;
    const int hi   = lane >> 4;         // lane half: selects K pair (ISA 16x4 f32 A layout)
    const int ln   = lane & 15;

    #pragma unroll
    for (int s = 0; s < 2; ++s) {
        // ternary selects on a fully unrolled loop: every load site sees a direct
        // kernarg pointer -> address space inferred global (no FLAT fallback)
        const float* W1  = s ? dnW1 : upW1;
        const float* b1  = s ? dnb1 : upb1;
        const float* W2  = s ? dnW2 : upW2;
        const float* b2  = s ? dnb2 : upb2;
        const float* pi_ = s ? dn_pi : up_pi;
        const float* sg_ = s ? dn_sg : up_sg;
        const float* fbase = features + ((size_t)b * N_EL + s * NSPIN) * D_MODEL;

        // ---------------- GEMM1: h = gelu(f @ W1 + b1), 32x256 ----------------
        // wave wv owns col-tiles jt = wv, wv+8 for BOTH row blocks in one K-loop:
        // A frags shared across col tiles, B frags shared across row blocks.
        {
            v8f acc00, acc01, acc10, acc11;    // [rowblk][coltile]
            const float bias0 = b1[wv * 16 + ln];
            const float bias1 = b1[(wv + 8) * 16 + ln];
            #pragma unroll
            for (int v = 0; v < 8; ++v) {
                acc00[v] = bias0; acc01[v] = bias1;
                acc10[v] = bias0; acc11[v] = bias1;
            }

            const float* arow0 = fbase + (size_t)(ln) * D_MODEL;        // rows 0..15
            const float* arow1 = fbase + (size_t)(16 + ln) * D_MODEL;   // rows 16..31

            for (int k0 = 0; k0 < D_MODEL; k0 += 4) {
                const int krow = k0 + hi * 2;
                // A 16x4 f32 frags: lane<16 holds K=k0,k0+1; lane>=16 holds K=k0+2,k0+3
                v2f a0 = *(const v2f*)(arow0 + krow);
                v2f a1 = *(const v2f*)(arow1 + krow);
                // B 4x16 f32 frags: VGPR v holds K=krow+v, cols across lanes
                v2f bA, bB;
                bA.x = W1[(size_t)krow * D_MODEL + wv * 16 + ln];
                bA.y = W1[(size_t)(krow + 1) * D_MODEL + wv * 16 + ln];
                bB.x = W1[(size_t)krow * D_MODEL + (wv + 8) * 16 + ln];
                bB.y = W1[(size_t)(krow + 1) * D_MODEL + (wv + 8) * 16 + ln];
                acc00 = __builtin_amdgcn_wmma_f32_16x16x4_f32(false, a0, false, bA,
                                                              (short)0, acc00, false, false);
                acc01 = __builtin_amdgcn_wmma_f32_16x16x4_f32(false, a0, false, bB,
                                                              (short)0, acc01, false, false);
                acc10 = __builtin_amdgcn_wmma_f32_16x16x4_f32(false, a1, false, bA,
                                                              (short)0, acc10, false, false);
                acc11 = __builtin_amdgcn_wmma_f32_16x16x4_f32(false, a1, false, bB,
                                                              (short)0, acc11, false, false);
            }
            // exact GELU + stage to LDS (C layout: VGPR v -> row v + hi*8, col = lane&15)
            #pragma unroll
            for (int v = 0; v < 8; ++v) {
                const int r0 = v + hi * 8;
                const int r1 = 16 + v + hi * 8;
                hbuf[r0][wv * 16 + ln]       = gelu_exact(acc00[v]);
                hbuf[r0][(wv + 8) * 16 + ln] = gelu_exact(acc01[v]);
                hbuf[r1][wv * 16 + ln]       = gelu_exact(acc10[v]);
                hbuf[r1][(wv + 8) * 16 + ln] = gelu_exact(acc11[v]);
            }
        }
        __syncthreads();

        // ---------- GEMM2 (waves 0-3): orb = h @ W2 + b2 ; wave 4: envelope ----------
        if (wv < 4) {
            const int i = wv >> 1, j = wv & 1;
            v8f acc;
            const float bias = b2[j * 16 + ln];
            #pragma unroll
            for (int v = 0; v < 8; ++v) acc[v] = bias;

            for (int k0 = 0; k0 < D_MODEL; k0 += 4) {
                const int krow = k0 + hi * 2;
                v2f a = *(const v2f*)&hbuf[i * 16 + ln][krow];
                v2f bb;
                bb.x = W2[(size_t)krow * NSPIN + j * 16 + ln];
                bb.y = W2[(size_t)(krow + 1) * NSPIN + j * 16 + ln];
                acc = __builtin_amdgcn_wmma_f32_16x16x4_f32(false, a, false, bb,
                                                            (short)0, acc, false, false);
            }
            #pragma unroll
            for (int v = 0; v < 8; ++v)
                orb[s][i * 16 + v + hi * 8][j * 16 + ln] = acc[v];
        } else if (wv == 4) {
            // envelope: lane = electron within this spin block
            const int e = s * NSPIN + lane;
            const float cx = ecoords[((size_t)b * N_EL + e) * 3 + 0];
            const float cy = ecoords[((size_t)b * N_EL + e) * 3 + 1];
            const float cz = ecoords[((size_t)b * N_EL + e) * 3 + 2];
            float env = 0.0f;
            #pragma unroll
            for (int a = 0; a < N_ATOM; ++a) {
                const float dx = cx - atom_coords[a * 3 + 0];
                const float dy = cy - atom_coords[a * 3 + 1];
                const float dz = cz - atom_coords[a * 3 + 2];
                const float r = sqrtf(dx * dx + dy * dy + dz * dz);
                env += pi_[a] * expf(-sg_[a] * r);
            }
            lenv[s][lane] = logf(fabsf(env));
        }
        __syncthreads();
    }

    // ---------------- LU with partial pivoting: wave 0 -> up, wave 1 -> down ----------------
    // log|det(diag(env) @ M)| = sum_n log|env_n| + sum_k log|U_kk|
    float ld = 0.0f;
    const bool act = (wv < 2);
    for (int k = 0; k < NSPIN; ++k) {
        if (act) {
            // pivot search down column k (lane = row)
            float v = (lane >= k) ? fabsf(orb[wv][lane][k]) : -1.0f;
            unsigned idx = (unsigned)lane;
            #pragma unroll
            for (int off = 16; off > 0; off >>= 1) {
                float ov = __shfl_xor(v, off, 32);
                unsigned oi = __shfl_xor(idx, off, 32);
                if (ov > v || (ov == v && oi < idx)) { v = ov; idx = oi; }
            }
            const int p = (int)idx;
            if (p != k) {                 // uniform branch; lane = column for the swap
                float t0 = orb[wv][k][lane];
                float t1 = orb[wv][p][lane];
                orb[wv][k][lane] = t1;
                orb[wv][p][lane] = t0;
            }
        }
        __syncthreads();
        if (act) {
            const float piv = orb[wv][k][k];
            ld += logf(fabsf(piv));       // every lane accumulates the same total
            if (lane > k) {               // lane = row, eliminate below pivot
                const float m = orb[wv][lane][k] / piv;
                for (int j = k + 1; j < NSPIN; ++j)
                    orb[wv][lane][j] -= m * orb[wv][k][j];
            }
        }
        __syncthreads();
    }

    if (act) {
        float ev = lenv[wv][lane];
        #pragma unroll
        for (int off = 16; off > 0; off >>= 1) ev += __shfl_xor(ev, off, 32);
        if (lane == 0) res[wv] = ld + ev;
    }
    __syncthreads();
    if (tid == 0) out[b] = res[0] + res[1];
}

extern "C" void kernel_launch(void* const* d_in, const int* in_sizes, int n_in,
                              void* d_out, int out_size, void* d_ws, size_t ws_size,
                              hipStream_t stream) {
    const float* features = (const float*)d_in[0];
    const float* ecoords  = (const float*)d_in[1];
    // d_in[2] = spins (layout is static: first 32 up, last 32 down) -> unused
    const float* atoms    = (const float*)d_in[3];
    const float* upW1 = (const float*)d_in[4];
    const float* upb1 = (const float*)d_in[5];
    const float* upW2 = (const float*)d_in[6];
    const float* upb2 = (const float*)d_in[7];
    const float* uppi = (const float*)d_in[8];
    const float* upsg = (const float*)d_in[9];
    const float* dnW1 = (const float*)d_in[10];
    const float* dnb1 = (const float*)d_in[11];
    const float* dnW2 = (const float*)d_in[12];
    const float* dnb2 = (const float*)d_in[13];
    const float* dnpi = (const float*)d_in[14];
    const float* dnsg = (const float*)d_in[15];
    float* out = (float*)d_out;

    const int B = in_sizes[0] / (N_EL * D_MODEL);   // 4096

    det_head_kernel<<<B, 256, 0, stream>>>(features, ecoords, atoms,
                                           upW1, upb1, upW2, upb2, uppi, upsg,
                                           dnW1, dnb1, dnW2, dnb2, dnpi, dnsg,
                                           out);
}